// VGAE_14199161881062
// MI455X (gfx1250) — compile-verified
//
#include <hip/hip_runtime.h>
#include <hip/hip_bf16.h>
#include <stdint.h>

typedef __attribute__((ext_vector_type(2))) float v2f;
typedef __attribute__((ext_vector_type(8))) float v8f;

#define NNODES 100000
#define NEDGES 1600000
#define EAD 7
#define HD 64
#define LAT 32
#define NGR 512
#define NCL 6
#define LDA 68   // padded LDS row stride (floats): bank = (r*4+k)%64, conflict-free

// ---------------------------------------------------------------------------
// Wave-level f32 WMMA tile: D(16x16) += A(16x64, LDS) * B(64xldb, global)
// using 16 chained V_WMMA_F32_16X16X4_F32 (K split in steps of 4).
// ---------------------------------------------------------------------------
__device__ __forceinline__ v8f wmma_tile(const float* __restrict__ Alds,
                                         const float* __restrict__ B,
                                         int ldb, int colbase, v8f acc)
{
    const int lane = threadIdx.x & 31;
    const int row  = lane & 15;          // A row (lanes 0-15 == 16-31 hold same M)
    const int kh   = (lane >> 4) << 1;   // lanes 16-31 carry K+2,K+3 halves
#pragma unroll
    for (int kk = 0; kk < HD; kk += 4) {
        v2f a, b;
        a.x = Alds[row * LDA + kk + kh];
        a.y = Alds[row * LDA + kk + kh + 1];
        b.x = B[(kk + kh)     * ldb + colbase + row];
        b.y = B[(kk + kh + 1) * ldb + colbase + row];
        acc = __builtin_amdgcn_wmma_f32_16x16x4_f32(false, a, false, b,
                                                    (short)0, acc, false, false);
    }
    return acc;
}

// ---------------------------------------------------------------------------
// RNG: counter hash + Box-Muller (stands in for jax threefry normal)
// ---------------------------------------------------------------------------
__device__ __forceinline__ uint32_t hashu(uint32_t x) {
    x ^= x >> 16; x *= 0x7feb352du;
    x ^= x >> 15; x *= 0x846ca68bu;
    x ^= x >> 16; return x;
}
__device__ __forceinline__ float gaussian_noise(uint32_t idx) {
    uint32_t r1 = hashu(idx * 2u + 0x9e3779b9u);
    uint32_t r2 = hashu(idx * 2u + 0x85ebca6bu);
    float u1 = (float)(r1 >> 8) * (1.0f / 16777216.0f) + (0.5f / 16777216.0f);
    float u2 = (float)(r2 >> 8) * (1.0f / 16777216.0f);
    float r  = sqrtf(-2.0f * __logf(u1));
    return r * __cosf(6.28318530718f * u2);
}

// ---------------------------------------------------------------------------
__global__ void k_zero(float* p, int n) {
    int i = blockIdx.x * blockDim.x + threadIdx.x;
    int s = gridDim.x * blockDim.x;
    for (; i < n; i += s) p[i] = 0.0f;
}

// conv1 edge phase: m = relu(x[src] + ea@We1 + be1); atomic scatter to agg1
__global__ void __launch_bounds__(256) k_conv1_edge(
    const float* __restrict__ x, const int* __restrict__ ei,
    const float* __restrict__ ea, const float* __restrict__ We1,
    const float* __restrict__ be1, float* __restrict__ agg1)
{
    int e = blockIdx.x * blockDim.x + threadIdx.x;
    if (e >= NEDGES) return;
    const int src = ei[e], dst = ei[NEDGES + e];
    float p = be1[0];
#pragma unroll
    for (int j = 0; j < EAD; ++j) p += ea[e * EAD + j] * We1[j];
    float m = fmaxf(x[src] + p, 0.0f);
    atomicAdd(&agg1[dst], m);
}

// conv1 node MLP: h1 = relu( relu(h*W11+b11) @ W12 + b12 ), h=(1+eps)x+agg
__global__ void __launch_bounds__(128) k_conv1_node(
    const float* __restrict__ x, const float* __restrict__ agg1,
    const float* __restrict__ W11, const float* __restrict__ b11,
    const float* __restrict__ W12, const float* __restrict__ b12,
    const float* __restrict__ eps1p, float* __restrict__ h1out)
{
    __shared__ float sHid[16 * LDA];
    const int tid = threadIdx.x;
    const int nb  = blockIdx.x * 16;
    const float e1 = 1.0f + eps1p[0];
    for (int i = tid; i < 16 * HD; i += 128) {
        int n = i >> 6, j = i & 63;
        int node = nb + n; if (node >= NNODES) node = NNODES - 1;
        float hv = e1 * x[node] + agg1[node];
        sHid[n * LDA + j] = fmaxf(hv * W11[j] + b11[j], 0.0f);
    }
    __syncthreads();
    const int lane = tid & 31, wave = tid >> 5;
    const int colbase = wave * 16;
    const int col = colbase + (lane & 15);
    v8f acc;
#pragma unroll
    for (int j = 0; j < 8; ++j) acc[j] = b12[col];
    acc = wmma_tile(sHid, W12, HD, colbase, acc);
    const int mbase = (lane >> 4) << 3;
#pragma unroll
    for (int j = 0; j < 8; ++j) {
        int node = nb + mbase + j;
        if (node < NNODES) h1out[node * HD + col] = fmaxf(acc[j], 0.0f);
    }
}

// conv2 edge phase: one wave per edge, lane -> float2 channel pair.
// Edge-uniform values forced scalar via readfirstlane.
__global__ void __launch_bounds__(256) k_conv2_edge(
    const float* __restrict__ h1, const int* __restrict__ ei,
    const float* __restrict__ ea, const float* __restrict__ We2,
    const float* __restrict__ be2, float* __restrict__ agg2)
{
    const int lane = threadIdx.x & 31;
    int e = blockIdx.x * 8 + (threadIdx.x >> 5);
    e = __builtin_amdgcn_readfirstlane(e);
    if (e >= NEDGES) return;
    const int src = ei[e], dst = ei[NEDGES + e];
    float pa[EAD];
#pragma unroll
    for (int j = 0; j < EAD; ++j) pa[j] = ea[e * EAD + j]; // scalar (uniform addr)
    const int c0 = lane << 1;
    float m0 = be2[c0], m1 = be2[c0 + 1];
#pragma unroll
    for (int j = 0; j < EAD; ++j) {
        m0 += pa[j] * We2[j * HD + c0];
        m1 += pa[j] * We2[j * HD + c0 + 1];
    }
    const float2 hv = *(const float2*)(h1 + (size_t)src * HD + c0); // coalesced 256B/wave
    m0 = fmaxf(hv.x + m0, 0.0f);
    m1 = fmaxf(hv.y + m1, 0.0f);
    atomicAdd(&agg2[(size_t)dst * HD + c0],     m0);
    atomicAdd(&agg2[(size_t)dst * HD + c0 + 1], m1);
}

// conv2 node MLP + VAE heads + reparameterize + pool, fused per 16-node tile
__global__ void __launch_bounds__(128) k_node2_fused(
    const float* __restrict__ h1, const float* __restrict__ agg2,
    const float* __restrict__ W21, const float* __restrict__ b21,
    const float* __restrict__ W22, const float* __restrict__ b22,
    const float* __restrict__ eps2p,
    const float* __restrict__ Wmu, const float* __restrict__ bmu,
    const float* __restrict__ Wlv, const float* __restrict__ blv,
    const int* __restrict__ batch,
    float* __restrict__ z_out, float* __restrict__ mu_out,
    float* __restrict__ lv_out, float* __restrict__ zsum,
    float* __restrict__ gcnt)
{
    __shared__ float sA[16 * LDA];
    __shared__ float sB[16 * LDA];
    const int tid = threadIdx.x;
    const int nb  = blockIdx.x * 16;
    const float e2 = 1.0f + eps2p[0];

    // t = (1+eps2)*h1 + agg2  -> sA
    for (int i = tid; i < 16 * HD; i += 128) {
        int n = i >> 6, c = i & 63;
        int node = nb + n; if (node >= NNODES) node = NNODES - 1;
        sA[n * LDA + c] = e2 * h1[(size_t)node * HD + c] + agg2[(size_t)node * HD + c];
    }
    __syncthreads();

    const int lane = tid & 31, wave = tid >> 5;
    const int mbase = (lane >> 4) << 3;

    { // u = relu(t @ W21 + b21)  -> sB
        const int colbase = wave * 16;
        const int col = colbase + (lane & 15);
        v8f acc;
#pragma unroll
        for (int j = 0; j < 8; ++j) acc[j] = b21[col];
        acc = wmma_tile(sA, W21, HD, colbase, acc);
#pragma unroll
        for (int j = 0; j < 8; ++j)
            sB[(mbase + j) * LDA + col] = fmaxf(acc[j], 0.0f);
    }
    __syncthreads();
    { // h2 = relu(u @ W22 + b22)  -> sA   (outer relu of conv2 folded in)
        const int colbase = wave * 16;
        const int col = colbase + (lane & 15);
        v8f acc;
#pragma unroll
        for (int j = 0; j < 8; ++j) acc[j] = b22[col];
        acc = wmma_tile(sB, W22, HD, colbase, acc);
#pragma unroll
        for (int j = 0; j < 8; ++j)
            sA[(mbase + j) * LDA + col] = fmaxf(acc[j], 0.0f);
    }
    __syncthreads();
    { // heads: waves 0,1 -> mu cols 0-31 ; waves 2,3 -> logvar cols 0-31
        const float* Wh = (wave < 2) ? Wmu : Wlv;   // wave-uniform select, EXEC stays full
        const float* bh = (wave < 2) ? bmu : blv;
        const int colbase = (wave & 1) * 16;
        const int col = colbase + (lane & 15);
        v8f acc;
#pragma unroll
        for (int j = 0; j < 8; ++j) acc[j] = bh[col];
        acc = wmma_tile(sA, Wh, LAT, colbase, acc);
        const int off = (wave >= 2) ? 32 : 0;       // mu -> sB[:,0:32], lv -> sB[:,32:64]
#pragma unroll
        for (int j = 0; j < 8; ++j)
            sB[(mbase + j) * LDA + off + col] = acc[j];
    }
    __syncthreads();

    // reparameterize + outputs + pooled sum
    for (int i = tid; i < 16 * LAT; i += 128) {
        int n = i >> 5, l = i & 31;
        int node = nb + n;
        if (node < NNODES) {
            float mu = sB[n * LDA + l];
            float lv = sB[n * LDA + 32 + l];
            float nz = gaussian_noise((uint32_t)(node * LAT + l));
            float zz = mu + nz * __expf(0.5f * lv);
            size_t o = (size_t)node * LAT + l;
            z_out[o]  = zz;
            mu_out[o] = mu;
            lv_out[o] = lv;
            atomicAdd(&zsum[(size_t)batch[node] * LAT + l], zz);
        }
    }
    if (tid < 16) {
        int node = nb + tid;
        if (node < NNODES) atomicAdd(&gcnt[batch[node]], 1.0f);
    }
}

// graph_emb = zsum / max(cnt,1); logits = emb @ Wc + bc
__global__ void __launch_bounds__(256) k_logits(
    const float* __restrict__ zsum, const float* __restrict__ gcnt,
    const float* __restrict__ Wc, const float* __restrict__ bc,
    float* __restrict__ out)
{
    int g = blockIdx.x * blockDim.x + threadIdx.x;
    if (g >= NGR) return;
    float inv = 1.0f / fmaxf(gcnt[g], 1.0f);
    float emb[LAT];
#pragma unroll
    for (int l = 0; l < LAT; ++l) emb[l] = zsum[g * LAT + l] * inv;
#pragma unroll
    for (int c = 0; c < NCL; ++c) {
        float s = bc[c];
#pragma unroll
        for (int l = 0; l < LAT; ++l) s += emb[l] * Wc[l * NCL + c];
        out[g * NCL + c] = s;
    }
}

// ---------------------------------------------------------------------------
extern "C" void kernel_launch(void* const* d_in, const int* in_sizes, int n_in,
                              void* d_out, int out_size, void* d_ws, size_t ws_size,
                              hipStream_t stream)
{
    const float* x     = (const float*)d_in[0];
    const int*   ei    = (const int*)  d_in[1];
    const float* ea    = (const float*)d_in[2];
    const int*   batch = (const int*)  d_in[3];
    const float* We1 = (const float*)d_in[4];
    const float* be1 = (const float*)d_in[5];
    const float* W11 = (const float*)d_in[6];
    const float* b11 = (const float*)d_in[7];
    const float* W12 = (const float*)d_in[8];
    const float* b12 = (const float*)d_in[9];
    const float* ep1 = (const float*)d_in[10];
    const float* We2 = (const float*)d_in[11];
    const float* be2 = (const float*)d_in[12];
    const float* W21 = (const float*)d_in[13];
    const float* b21 = (const float*)d_in[14];
    const float* W22 = (const float*)d_in[15];
    const float* b22 = (const float*)d_in[16];
    const float* ep2 = (const float*)d_in[17];
    const float* Wmu = (const float*)d_in[18];
    const float* bmu = (const float*)d_in[19];
    const float* Wlv = (const float*)d_in[20];
    const float* blv = (const float*)d_in[21];
    const float* Wc  = (const float*)d_in[22];
    const float* bc  = (const float*)d_in[23];

    // workspace: contiguous zero-init region first, h1 after it
    float* ws   = (float*)d_ws;
    float* agg1 = ws;                                   // N
    float* agg2 = agg1 + NNODES;                        // N*H
    float* zsum = agg2 + (size_t)NNODES * HD;           // G*LAT
    float* gcnt = zsum + NGR * LAT;                     // G
    float* h1   = gcnt + NGR;                           // N*H
    const int zeroCount = NNODES + NNODES * HD + NGR * LAT + NGR;

    float* z_out  = (float*)d_out;
    float* mu_out = z_out  + (size_t)NNODES * LAT;
    float* lv_out = mu_out + (size_t)NNODES * LAT;
    float* lg_out = lv_out + (size_t)NNODES * LAT;

    k_zero<<<2048, 256, 0, stream>>>(ws, zeroCount);
    k_conv1_edge<<<(NEDGES + 255) / 256, 256, 0, stream>>>(x, ei, ea, We1, be1, agg1);
    k_conv1_node<<<NNODES / 16, 128, 0, stream>>>(x, agg1, W11, b11, W12, b12, ep1, h1);
    k_conv2_edge<<<NEDGES / 8, 256, 0, stream>>>(h1, ei, ea, We2, be2, agg2);
    k_node2_fused<<<NNODES / 16, 128, 0, stream>>>(h1, agg2, W21, b21, W22, b22, ep2,
                                                   Wmu, bmu, Wlv, blv, batch,
                                                   z_out, mu_out, lv_out, zsum, gcnt);
    k_logits<<<2, 256, 0, stream>>>(zsum, gcnt, Wc, bc, lg_out);
}